// MostPopularEncoder_70935679861351
// MI455X (gfx1250) — compile-verified
//
#include <hip/hip_runtime.h>

// MostPopularEncoder: avg_delta = (ts[t]-ts[0])/(t+1); running mode of labels.
// Mode-scan reduced to an associative running-max of key=(occ_rank<<10)|(1023-label).
// One wave32 per sequence; ballot-based multi-match for in-tile occurrence ranks;
// LDS histogram updated with one ds_add_rtn_u32 per distinct label per 32-step tile.

#define B_ 32
#define L_ 2048
#define TILES_ (L_ / 32)

__global__ __launch_bounds__(32)
void mode_scan_kernel(const float* __restrict__ ts,
                      const int* __restrict__ labels,
                      float* __restrict__ out) {
    __shared__ int cnt[1024];  // per-class running counts for this sequence

    const int lane = threadIdx.x;          // 0..31, wave32
    const int b    = blockIdx.x;           // sequence id
    const size_t rowOff = (size_t)b * L_;  // element offset of this row

    // zero histogram
    for (int i = lane; i < 1024; i += 32) cnt[i] = 0;
    __syncthreads();

    const float ts0 = ts[rowOff];          // uniform -> scalar load
    int carry = -1;                        // running-max key from previous tiles

    for (int tile = 0; tile < TILES_; ++tile) {
        const int   t   = tile * 32 + lane;
        const int   lab = labels[rowOff + t];   // coalesced 128B/wave
        const float tsv = ts[rowOff + t];       // coalesced 128B/wave

        // prefetch next tile (global_prefetch_b8 on gfx1250)
        if (tile + 1 < TILES_) {
            __builtin_prefetch(&labels[rowOff + t + 32], 0, 0);
            __builtin_prefetch(&ts[rowOff + t + 32], 0, 0);
        }

        // --- multi-match: mask of lanes holding the same 10-bit label ---
        unsigned mask = 0xffffffffu;
        #pragma unroll
        for (int bit = 0; bit < 10; ++bit) {
            unsigned vote = (unsigned)__ballot((lab >> bit) & 1);
            mask &= ((lab >> bit) & 1) ? vote : ~vote;
        }
        const unsigned below  = (1u << lane) - 1u;
        const int      rank   = __popc(mask & below);  // in-tile occurrence rank
        const int      total  = __popc(mask);          // occurrences of lab in tile
        const int      leader = __ffs(mask) - 1;       // lowest lane with this label

        // one LDS atomic per distinct label in the tile; broadcast the base count
        int base = 0;
        if (rank == 0) base = atomicAdd(&cnt[lab], total);  // ds_add_rtn_u32
        base = __shfl(base, leader);

        // global occurrence rank (time-ordered), packed key
        const int r   = base + rank + 1;            // <= 2048, fits key[21:10]
        int       key = (r << 10) | (1023 - lab);   // max r, ties -> min label

        // --- inclusive max-scan across the wave (5 shuffle steps) ---
        #pragma unroll
        for (int off = 1; off < 32; off <<= 1) {
            int y = __shfl_up(key, off);
            if (lane >= off) key = (y > key) ? y : key;
        }
        key = (carry > key) ? carry : key;
        carry = __shfl(key, 31);

        const int   top = 1023 - (key & 1023);
        const float avg = (tsv - ts0) / (float)(t + 1);

        // hiddens[b,t,:] = {avg_delta, top_label}; tuple output -> write twice
        float2 v = make_float2(avg, (float)top);
        float2* o = (float2*)out;
        o[rowOff + t] = v;                          // hiddens        [B,L,2]
        o[(size_t)B_ * L_ + rowOff + t] = v;        // hiddens[None] [1,B,L,2]
    }
}

extern "C" void kernel_launch(void* const* d_in, const int* in_sizes, int n_in,
                              void* d_out, int out_size, void* d_ws, size_t ws_size,
                              hipStream_t stream) {
    (void)in_sizes; (void)n_in; (void)d_ws; (void)ws_size; (void)out_size;
    const float* ts     = (const float*)d_in[0];  // timestamps [32,2048] f32
    const int*   labels = (const int*)d_in[1];    // labels     [32,2048] i32
    // d_in[2] = num_classes (==1000), compile-time constant here
    float* out = (float*)d_out;                   // 2*32*2048*2 f32
    mode_scan_kernel<<<dim3(B_), dim3(32), 0, stream>>>(ts, labels, out);
}